// lossLayer_89979564851725
// MI455X (gfx1250) — compile-verified
//
#include <hip/hip_runtime.h>
#include <stdint.h>

// ---------------------------------------------------------------------------
// Loss layer: BS=2048 batches, P=32 grid, C=14 channels.
// HBM-bound (117 MB in / 4 B out). Strategy: async global->LDS double-buffered
// streaming (CDNA5 ASYNCcnt path, via inline asm global_load_async_to_lds_b128),
// per-thread (j,c) column products, small LDS tree for cross-column products,
// deterministic two-kernel reduction.
// ---------------------------------------------------------------------------

#define BS    2048
#define PDIM  32
#define CCH   14
#define NT    (PDIM * CCH)          // 448 threads: thread t -> (j = t/14, c = t%14)
#define PERB  (PDIM * PDIM * CCH)   // 14336 floats per batch
#define HALF  (PERB / 2)            // 7168 floats per half-batch stage (16 rows)
#define NB    8                     // batches per block
#define NBLK  (BS / NB)             // 256 blocks
#define NSTG  (2 * NB)              // pipeline stages per block
#define CHUNK 4                     // 16-byte async copies per thread per stage

// Async global->LDS copy, GV mode (CDNA5 ISA):
//   VDST = per-lane LDS byte address, VADDR = 64-bit global address.
// LDS byte address = low 32 bits of the generic pointer (LDS aperture maps
// addr[31:0] to the LDS offset). Inline asm bypasses the builtin's
// CUDA-address-space pointer typing, which HIP source cannot spell.
__device__ __forceinline__ void async_copy_b128(const float* gsrc, float* ldst) {
  uint32_t ldsoff = (uint32_t)(uintptr_t)ldst;
  asm volatile("global_load_async_to_lds_b128 %0, %1, off"
               :: "v"(ldsoff), "v"(gsrc) : "memory");
}

#if __has_builtin(__builtin_amdgcn_s_wait_asynccnt)
#define ASYNC_WAIT(n) __builtin_amdgcn_s_wait_asynccnt(n)
#else
#define ASYNC_WAIT(n) asm volatile("s_wait_asynccnt %0" :: "i"(n) : "memory")
#endif

__global__ __launch_bounds__(NT) void loss_main(
    const float* __restrict__ y_pred,
    const float* __restrict__ eta,
    const float* __restrict__ p,
    const float* __restrict__ bbox,
    float* __restrict__ partial)
{
  __shared__ __align__(16) float sbuf[2][HALF]; // double-buffered tiles (2 x 28 KB)
  __shared__ float red[3][NT];      // per-(j,c) partials: all / masked-yp / masked-tmp
  __shared__ float chan[3][CCH];    // per-channel products
  __shared__ float lsum[CCH];       // per-channel loss terms

  const int t  = threadIdx.x;
  const int j  = t / CCH;
  const int c  = t - j * CCH;
  const int b0 = blockIdx.x * NB;

  const float* ybase = y_pred + (size_t)b0 * PERB;

  // Prologue: issue stage 0 (batch b0, rows 0..15) into sbuf[0].
#pragma unroll
  for (int k = 0; k < CHUNK; ++k) {
    const int off = (t + k * NT) * 4;   // 16B-chunk float offset
    async_copy_b128(ybase + off, &sbuf[0][off]);
  }

  float acc = 0.0f;
  int x0 = 0, y0 = 0, ww = 0, hh = 0;
  bool col_in = false;
  float a_all = 1.0f, a_ypm = 1.0f, a_tmm = 1.0f;

  for (int s = 0; s < NSTG; ++s) {
    const int cur = s & 1;            // buffer parity == half index
    const int n   = s >> 1;           // batch within block

    // Prefetch next stage into the other buffer, then wait for current stage.
    // Async loads complete in order: waiting ASYNCcnt<=CHUNK guarantees the
    // CHUNK earlier (current-stage) copies are done while the prefetch flies.
    if (s + 1 < NSTG) {
      const float* g = ybase + (size_t)(s + 1) * HALF;
      float* db = sbuf[cur ^ 1];
#pragma unroll
      for (int k = 0; k < CHUNK; ++k) {
        const int off = (t + k * NT) * 4;
        async_copy_b128(g + off, db + off);
      }
      ASYNC_WAIT(CHUNK);
    } else {
      ASYNC_WAIT(0);
    }
    __syncthreads();                  // current tile visible to all waves

    if (cur == 0) {
      // New batch: load bbox window for (b, c), reset product accumulators.
      const int b = b0 + n;
      const float* bb = bbox + ((size_t)b * CCH + c) * 4;
      x0 = (int)(bb[0] * (float)PDIM);
      y0 = (int)(bb[1] * (float)PDIM);
      ww = (int)(bb[2] * (float)PDIM);
      hh = (int)(bb[3] * (float)PDIM);
      col_in = (j >= x0) && (j < x0 + ww);
      a_all = 1.0f; a_ypm = 1.0f; a_tmm = 1.0f;
    }

    // Column walk: 16 rows of this stage. LDS dword index i_local*448 + t ->
    // bank t%64, conflict-free within each wave.
    const float* src  = &sbuf[cur][t];
    const int ibase = cur * (PDIM / 2);
#pragma unroll
    for (int il = 0; il < PDIM / 2; ++il) {
      const int i = ibase + il;
      const float v  = src[il * NT];
      const float yc = fminf(1.0f, v);
      const float tr = fminf(1.0f, 1.0f - yc);
      const float yp = __builtin_fmaf(yc, 0.02f, 0.98f);
      const float tm = __builtin_fmaf(tr, 0.02f, 0.98f);
      a_all *= tm;
      const bool row_in = (i >= y0) && (i < y0 + hh);
      a_ypm *= row_in ? yp : 1.0f;
      a_tmm *= row_in ? tm : 1.0f;
    }

    if (cur == 1) {
      // Batch complete: fold 32 columns per channel, then the loss.
      red[0][t] = a_all;
      red[1][t] = col_in ? a_ypm : 1.0f;
      red[2][t] = col_in ? a_tmm : 1.0f;
      __syncthreads();

      if (t < 3 * CCH) {
        const int which = t / CCH;
        const int cc    = t - which * CCH;
        float prod = 1.0f;
#pragma unroll
        for (int jj = 0; jj < PDIM; ++jj) prod *= red[which][jj * CCH + cc];
        chan[which][cc] = prod;
      }
      __syncthreads();

      if (t < CCH) {
        const int b = b0 + n;
        const float pt   = chan[0][t];          // prod_tmp (all cells)
        const float p1m  = chan[1][t];          // prod yp inside bbox
        const float p3m  = chan[2][t];          // prod tmp inside bbox
        const float pyx  = 1.0f - pt;
        const float pbb  = p1m * pt / p3m + 1e-10f;
        const float e    = eta[(size_t)b * CCH + t];
        const float pr   = p[(size_t)b * CCH + t];
        lsum[t] = -e * logf(pbb)
                  - (1.0f - e) * pr * logf(pyx)
                  - (1.0f - e) * (1.0f - pr) * logf(1.0f - pyx);
      }
      __syncthreads();

      if (t == 0) {
        float sL = 0.0f;
#pragma unroll
        for (int cc = 0; cc < CCH; ++cc) sL += lsum[cc];
        acc += sL;
      }
    }
    __syncthreads();   // protects sbuf[cur] before stage s+1 overwrites it
  }

  if (t == 0) partial[blockIdx.x] = acc;
}

__global__ __launch_bounds__(NBLK) void loss_finalize(
    const float* __restrict__ partial, float* __restrict__ out)
{
  __shared__ float s[NBLK];
  const int t = threadIdx.x;
  s[t] = partial[t];
  __syncthreads();
#pragma unroll
  for (int st = NBLK / 2; st > 0; st >>= 1) {
    if (t < st) s[t] += s[t + st];
    __syncthreads();
  }
  if (t == 0) out[0] = s[0] / (float)BS;
}

extern "C" void kernel_launch(void* const* d_in, const int* in_sizes, int n_in,
                              void* d_out, int out_size, void* d_ws, size_t ws_size,
                              hipStream_t stream) {
  (void)in_sizes; (void)n_in; (void)out_size; (void)ws_size;
  const float* y_pred = (const float*)d_in[0];
  const float* eta    = (const float*)d_in[1];
  const float* p      = (const float*)d_in[2];
  const float* bbox   = (const float*)d_in[3];
  float* partial = (float*)d_ws;          // NBLK floats of scratch
  float* out     = (float*)d_out;

  hipLaunchKernelGGL(loss_main, dim3(NBLK), dim3(NT), 0, stream,
                     y_pred, eta, p, bbox, partial);
  hipLaunchKernelGGL(loss_finalize, dim3(1), dim3(NBLK), 0, stream,
                     partial, out);
}